// Sparse_54855322305130
// MI455X (gfx1250) — compile-verified
//
#include <hip/hip_runtime.h>
#include <hip/hip_bf16.h>

typedef float v2f  __attribute__((ext_vector_type(2)));
typedef float v8f  __attribute__((ext_vector_type(8)));
typedef unsigned int u32x4 __attribute__((ext_vector_type(4)));
typedef unsigned int u32x8 __attribute__((ext_vector_type(8)));

#define N_FEAT 4096
#define IN_DIM 4096
#define BATCH  4096

#define TILE 128     // output tile per workgroup (M and N)
#define BK   32      // K-slab depth staged in LDS
#define LDA  34      // LDS row stride in floats = 32 + TDM pad (2 DW per 32 DW)

// ---------------------------------------------------------------------------
// Stage 1: COO scatter-add densify: Kd[row*IN_DIM + col] += val
// ---------------------------------------------------------------------------
__global__ __launch_bounds__(256) void coo_densify_kernel(
    const float* __restrict__ vals, const int* __restrict__ rows,
    const int* __restrict__ cols, float* __restrict__ Kd, int nnz) {
  int i = blockIdx.x * blockDim.x + threadIdx.x;
  if (i < nnz) {
    size_t off = (size_t)rows[i] * IN_DIM + (size_t)cols[i];
    atomicAdd(&Kd[off], vals[i]);  // global_atomic_add_f32 (duplicates sum)
  }
}

// ---------------------------------------------------------------------------
// TDM: issue one tensor_load_to_lds for a 128x32 fp32 tile (row stride
// IN_DIM), with TDM-side LDS padding of 2 DWORDs every 32 DWORDs so each
// 32-float row lands at a 34-float LDS stride (bank-conflict-free b64 reads).
// Descriptor layout per CDNA5 ISA.
// ---------------------------------------------------------------------------
__device__ __forceinline__ void tdm_load_tile_128x32(const float* gsrc,
                                                     unsigned lds_byte_off) {
  const unsigned long long ga = (unsigned long long)(uintptr_t)gsrc;
  u32x4 g0;
  g0.x = 1u;                                       // count=1, is_restore=0, gather=0
  g0.y = lds_byte_off;                             // lds_addr (bytes)
  g0.z = (unsigned)(ga & 0xFFFFFFFFu);             // global_addr[31:0]
  g0.w = (unsigned)((ga >> 32) & 0x01FFFFFFu)      // global_addr[56:32]
         | (2u << 30);                             // type=2 ("image")
  u32x8 g1;
  g1[0] = (2u << 16)                               // data_size = 4 bytes
        | (1u << 20)                               // pad_enable
        | (4u << 22)                               // pad_interval: 32 DWORDs
        | (1u << 25);                              // pad_amount:   2 DWORDs
  g1[1] = (4096u << 16);                           // abar_addr=0 | tensor_dim0 lo16
  g1[2] = (4096u << 16);                           // tensor_dim0 hi=0 | tensor_dim1 lo16
  g1[3] = (32u << 16);                             // tensor_dim1 hi=0 | tile_dim0=32
  g1[4] = 128u;                                    // tile_dim1=128 | tile_dim2=0
  g1[5] = (unsigned)IN_DIM;                        // tensor_dim0_stride lo32 = 4096
  g1[6] = 0u;                                      // stride0 hi | dim1_stride lo (2D: unused)
  g1[7] = 0u;
  asm volatile("tensor_load_to_lds %0, %1" :: "s"(g0), "s"(g1) : "memory");
}

// ---------------------------------------------------------------------------
// Stage 2: C[b,f] = sum_d X[b,d] * Kd[f,d] + bias[f]
// Double-buffered TDM -> LDS, fp32 WMMA GEMM. 8 waves (wave32), 128x128 tile.
// Wave grid 2(M) x 4(N); each wave: 4x2 tiles of 16x16 -> 64x32 of C.
// ---------------------------------------------------------------------------
__global__ __launch_bounds__(256) void coo_gemm_kernel(
    const float* __restrict__ X,     // [BATCH][IN_DIM]
    const float* __restrict__ Kd,    // [N_FEAT][IN_DIM] densified
    const float* __restrict__ bias,  // [N_FEAT]
    float* __restrict__ C) {         // [BATCH][N_FEAT]
  __shared__ __align__(16) float Abuf[2][TILE * LDA];  // X  tiles: [m][k]
  __shared__ __align__(16) float Bbuf[2][TILE * LDA];  // Kd tiles: [n][k]

  const int t    = threadIdx.x;
  const int wave = t >> 5;
  const int lane = t & 31;
  const int wm   = wave >> 2;       // 0..1  (M sub-block of 64)
  const int wn   = wave & 3;        // 0..3  (N sub-block of 32)
  const int half = lane >> 4;       // 0/1 -> which K-pair / which M-half
  const int l16  = lane & 15;

  const int bm0 = blockIdx.y * TILE;   // batch rows
  const int fn0 = blockIdx.x * TILE;   // feature cols

  v8f acc[4][2];
#pragma unroll
  for (int mi = 0; mi < 4; ++mi)
#pragma unroll
    for (int nj = 0; nj < 2; ++nj) acc[mi][nj] = (v8f)0.0f;

  // LDS byte offsets of the four staging buffers (low 32 bits of flat addr).
  const unsigned aoff0 = (unsigned)(uintptr_t)&Abuf[0][0];
  const unsigned aoff1 = (unsigned)(uintptr_t)&Abuf[1][0];
  const unsigned boff0 = (unsigned)(uintptr_t)&Bbuf[0][0];
  const unsigned boff1 = (unsigned)(uintptr_t)&Bbuf[1][0];

  const int NSLAB = IN_DIM / BK;  // 128

  // Prologue: TDM-load slab 0 into buffer 0 (issued once per workgroup).
  if (wave == 0) {
    tdm_load_tile_128x32(X  + (size_t)bm0 * IN_DIM, aoff0);
    tdm_load_tile_128x32(Kd + (size_t)fn0 * IN_DIM, boff0);
  }

  for (int i = 0; i < NSLAB; ++i) {
    const int p = i & 1;
    __syncthreads();  // all waves done computing on buffer p^1 (slab i-1)
    if (wave == 0) {
      if (i + 1 < NSLAB) {
        const int kb2 = (i + 1) * BK;
        tdm_load_tile_128x32(X  + (size_t)bm0 * IN_DIM + kb2, p ? aoff0 : aoff1);
        tdm_load_tile_128x32(Kd + (size_t)fn0 * IN_DIM + kb2, p ? boff0 : boff1);
        __builtin_amdgcn_s_wait_tensorcnt(2);  // in-order: slab i's 2 ops done
      } else {
        __builtin_amdgcn_s_wait_tensorcnt(0);
      }
    }
    __syncthreads();  // buffer p now holds slab i

    const float* As = &Abuf[p][0];
    const float* Bs = &Bbuf[p][0];
#pragma unroll
    for (int kk = 0; kk < BK; kk += 4) {
      const int ko = kk + half * 2;  // lanes 0-15: K=kk,kk+1 ; lanes 16-31: K=kk+2,kk+3
      v2f a[4], b[2];
#pragma unroll
      for (int mi = 0; mi < 4; ++mi)
        a[mi] = *(const v2f*)(As + (wm * 64 + mi * 16 + l16) * LDA + ko);
#pragma unroll
      for (int nj = 0; nj < 2; ++nj)
        b[nj] = *(const v2f*)(Bs + (wn * 32 + nj * 16 + l16) * LDA + ko);
#pragma unroll
      for (int mi = 0; mi < 4; ++mi)
#pragma unroll
        for (int nj = 0; nj < 2; ++nj)
          acc[mi][nj] = __builtin_amdgcn_wmma_f32_16x16x4_f32(
              /*neg_a=*/false, a[mi], /*neg_b=*/false, b[nj],
              /*c_mod=*/(short)0, acc[mi][nj],
              /*reuse_a=*/false, /*reuse_b=*/false);
    }
  }

  // Epilogue: C/D layout -> lane l16 = N, VGPR r = M row (r + 8*half), fuse bias.
#pragma unroll
  for (int nj = 0; nj < 2; ++nj) {
    const int col = fn0 + wn * 32 + nj * 16 + l16;
    const float bv = bias[col];
#pragma unroll
    for (int mi = 0; mi < 4; ++mi) {
#pragma unroll
      for (int r = 0; r < 8; ++r) {
        const int row = bm0 + wm * 64 + mi * 16 + half * 8 + r;
        C[(size_t)row * N_FEAT + col] = acc[mi][nj][r] + bv;
      }
    }
  }
}

// ---------------------------------------------------------------------------
extern "C" void kernel_launch(void* const* d_in, const int* in_sizes, int n_in,
                              void* d_out, int out_size, void* d_ws, size_t ws_size,
                              hipStream_t stream) {
  const float* X    = (const float*)d_in[0];   // [BATCH*IN_DIM]
  const float* vals = (const float*)d_in[1];   // [nnz]
  const int*   rows = (const int*)d_in[2];     // [nnz]
  const int*   cols = (const int*)d_in[3];     // [nnz]
  const float* bias = (const float*)d_in[4];   // [N_FEAT]
  float*       out  = (float*)d_out;           // [BATCH*N_FEAT]
  float*       Kd   = (float*)d_ws;            // dense kernel scratch (64 MB)

  const int nnz = in_sizes[1];
  const size_t kbytes = (size_t)N_FEAT * IN_DIM * sizeof(float);

  // Zero dense kernel (graph-capturable), scatter-add COO, then WMMA GEMM.
  hipMemsetAsync(Kd, 0, kbytes, stream);
  coo_densify_kernel<<<(nnz + 255) / 256, 256, 0, stream>>>(vals, rows, cols, Kd, nnz);

  dim3 grid(N_FEAT / TILE, BATCH / TILE);  // 32 x 32
  coo_gemm_kernel<<<grid, 256, 0, stream>>>(X, Kd, bias, out);
}